// HeteroSAGEConv_59854664237622
// MI455X (gfx1250) — compile-verified
//
#include <hip/hip_runtime.h>
#include <hip/hip_bf16.h>

// ---------------------------------------------------------------------------
// HeteroSAGEConv for MI455X (gfx1250, wave32, WMMA + TDM)
//
// Sizes (fixed by reference): N_USER = N_ITEM = 50000, E = 600000, D = H = 128
// ---------------------------------------------------------------------------

typedef float v2f __attribute__((ext_vector_type(2)));
typedef float v8f __attribute__((ext_vector_type(8)));
typedef unsigned int v4u __attribute__((ext_vector_type(4)));
typedef unsigned int v8u __attribute__((ext_vector_type(8)));

#define N_NODES 50000
#define N_EDGES 600000
#define DIMH    128

// ---------------------------------------------------------------------------
// Tensor Data Mover: DMA the 128x128 f32 weight matrix from global into LDS.
// D# per CDNA5 ISA:
//   group0: count=1 | lds_addr | global_addr[56:0] | type=2
//   group1: data_size=4B, tensor_dim0/1 = 128, tile_dim0/1 = 128,
//           tensor_dim0_stride = 128 (elements), 2D tile (tile_dim2 = 0)
// 2-group form (VADDR2/VADDR3 = NULL) is legal for tensors up to 2D.
// Completion tracked with TENSORcnt.
// ---------------------------------------------------------------------------
__device__ __forceinline__ void tdm_load_w_to_lds(const float* W, float* sW)
{
    const unsigned long long ga = (unsigned long long)(uintptr_t)W;
    // Generic (flat) LDS pointer: addr[31:0] is the LDS byte offset.
    const unsigned lds_off = (unsigned)(uintptr_t)sW;

    v4u g0;
    g0.x = 1u;                                                 // count=1 (valid user D#)
    g0.y = lds_off;                                            // lds_addr
    g0.z = (unsigned)(ga & 0xffffffffu);                       // global_addr[31:0]
    g0.w = (unsigned)((ga >> 32) & 0x01ffffffu) | (2u << 30);  // global_addr[56:32] | type=2

    v8u g1;
    g1[0] = 2u << 16;                  // workgroup_mask=0, data_size=2 (4 bytes)
    g1[1] = (unsigned)DIMH << 16;      // tensor_dim0 = 128 (bits 79:48, low half)
    g1[2] = (unsigned)DIMH << 16;      // tensor_dim0 hi=0 | tensor_dim1 = 128 (low half)
    g1[3] = (unsigned)DIMH << 16;      // tensor_dim1 hi=0 | tile_dim0 = 128
    g1[4] = (unsigned)DIMH;            // tile_dim1 = 128 | tile_dim2 = 0
    g1[5] = (unsigned)DIMH;            // tensor_dim0_stride = 128 (bits 207:160, low)
    g1[6] = (unsigned)(DIMH * DIMH) << 16;  // dim0_stride hi=0 | tensor_dim1_stride low
    g1[7] = 0u;                        // tensor_dim1_stride hi

    asm volatile("tensor_load_to_lds %0, %1"
                 :
                 : "s"(g0), "s"(g1)
                 : "memory");
    __builtin_amdgcn_s_wait_tensorcnt(0);
}

// ---------------------------------------------------------------------------
// GEMM: Y[N,128] = X[N,128] @ W[128,128]   (optionally Y += ...)
// Block: 256 threads = 8 waves. Each block computes a 128-row stripe.
// Wave w computes rows [row0+16w, row0+16w+16) x all 128 columns using
// eight f32 16x16x4 WMMA accumulators (K-loop of 32 steps).
// W (64 KB) is DMA'd into LDS once per block by wave 0 (TDM) and shared.
// ---------------------------------------------------------------------------
__global__ void __launch_bounds__(256)
gemm_proj_kernel(const float* __restrict__ X, const float* __restrict__ W,
                 float* __restrict__ Y, int N, int accumulate)
{
    __shared__ float sW[DIMH * DIMH];   // 64 KB

    const int tid = threadIdx.x;

    // Async tensor DMA of W into LDS; wave 0 issues + waits, then block syncs.
    if (tid < 32) {
        tdm_load_w_to_lds(W, sW);
    }
    __syncthreads();

    const int wave = tid >> 5;
    const int lane = tid & 31;
    const int half = lane >> 4;     // 0: lanes 0-15, 1: lanes 16-31
    const int l16  = lane & 15;

    const int row0 = blockIdx.x * 128 + wave * 16;

    // A-fragment source row (clamped for the ragged last block; stores are
    // guarded so clamped rows never write).
    int arow = row0 + l16;
    if (arow >= N) arow = N - 1;
    if (arow < 0)  arow = 0;
    const float* __restrict__ xrow = X + (size_t)arow * DIMH;

    v8f acc[8];
#pragma unroll
    for (int n = 0; n < 8; ++n) {
        v8f z = {0.f, 0.f, 0.f, 0.f, 0.f, 0.f, 0.f, 0.f};
        acc[n] = z;
    }

    // K loop: 32 steps of 16x16x4.
    // A layout (ISA 7.12.2): lanes 0-15 hold (K=k0,K=k0+1) of row M=l16,
    //                        lanes 16-31 hold (K=k0+2,K=k0+3) of row M=l16.
    // B layout (analogous):  VGPR0 = row K (lanes 0-15) / K+2 (lanes 16-31),
    //                        VGPR1 = row K+1 / K+3, column N = l16 + 16n.
    for (int k0 = 0; k0 < DIMH; k0 += 4) {
        const int kk = k0 + (half << 1);
        v2f a;
        a.x = xrow[kk];
        a.y = xrow[kk + 1];
#pragma unroll
        for (int n = 0; n < 8; ++n) {
            const int col = (n << 4) + l16;
            v2f b;
            b.x = sW[kk * DIMH + col];
            b.y = sW[(kk + 1) * DIMH + col];
            acc[n] = __builtin_amdgcn_wmma_f32_16x16x4_f32(
                /*neg_a=*/false, a, /*neg_b=*/false, b,
                /*c_mod=*/(short)0, acc[n],
                /*reuse_a=*/false, /*reuse_b=*/false);
        }
    }

    // D layout: element r of acc[n] in lane L -> row = row0 + r + 8*half,
    //           col = 16n + (L&15).
    const int srow = row0 + (half << 3);
    if (row0 + 16 <= N) {
        // Fast path: whole wave-tile in range (all blocks but the last).
        if (accumulate) {
#pragma unroll
            for (int n = 0; n < 8; ++n) {
                const int col = (n << 4) + l16;
#pragma unroll
                for (int r = 0; r < 8; ++r)
                    Y[(size_t)(srow + r) * DIMH + col] += acc[n][r];
            }
        } else {
#pragma unroll
            for (int n = 0; n < 8; ++n) {
                const int col = (n << 4) + l16;
#pragma unroll
                for (int r = 0; r < 8; ++r)
                    Y[(size_t)(srow + r) * DIMH + col] = acc[n][r];
            }
        }
    } else {
#pragma unroll
        for (int n = 0; n < 8; ++n) {
            const int col = (n << 4) + l16;
#pragma unroll
            for (int r = 0; r < 8; ++r) {
                const int rr = srow + r;
                if (rr < N) {
                    const size_t o = (size_t)rr * DIMH + col;
                    const float v = acc[n][r];
                    if (accumulate) Y[o] += v;
                    else            Y[o]  = v;
                }
            }
        }
    }
}

// ---------------------------------------------------------------------------
// Scatter-add: one wave (32 lanes) per edge, float4 per lane (128 floats),
// native f32 atomics into the aggregation buffer; lane 0 bumps the count.
// ---------------------------------------------------------------------------
__global__ void __launch_bounds__(256)
scatter_add_kernel(const float* __restrict__ srcproj,
                   const int* __restrict__ esrc, const int* __restrict__ etgt,
                   int E, float* __restrict__ agg, float* __restrict__ cnt)
{
    const int g    = blockIdx.x * blockDim.x + threadIdx.x;
    const int edge = g >> 5;
    const int lane = g & 31;
    if (edge >= E) return;

    const int s = esrc[edge];
    const int t = etgt[edge];

    const float4 v = ((const float4*)(srcproj + (size_t)s * DIMH))[lane];
    float* __restrict__ dst = agg + (size_t)t * DIMH + (lane << 2);
    unsafeAtomicAdd(dst + 0, v.x);
    unsafeAtomicAdd(dst + 1, v.y);
    unsafeAtomicAdd(dst + 2, v.z);
    unsafeAtomicAdd(dst + 3, v.w);
    if (lane == 0) unsafeAtomicAdd(cnt + t, 1.0f);
}

// ---------------------------------------------------------------------------
// Combine kernels: mean-divide per relation, sum, scale, double-ReLU (=ReLU).
// One thread per float4 (32 threads per node row).
// ---------------------------------------------------------------------------
__global__ void __launch_bounds__(256)
combine_user_kernel(const float* __restrict__ tgt_user,
                    const float* __restrict__ agg_rated,
                    const float* __restrict__ cnt_rated,
                    const float* __restrict__ agg_follows,
                    const float* __restrict__ cnt_follows,
                    float* __restrict__ out_user, int n)
{
    const int g = blockIdx.x * blockDim.x + threadIdx.x;
    if (g >= n * 32) return;
    const int node = g >> 5;
    const float cr = fmaxf(cnt_rated[node], 1.0f);
    const float cf = fmaxf(cnt_follows[node], 1.0f);
    const float4 t  = ((const float4*)tgt_user)[g];
    const float4 ar = ((const float4*)agg_rated)[g];
    const float4 af = ((const float4*)agg_follows)[g];
    float4 o;
    o.x = fmaxf((t.x + ar.x / cr + af.x / cf) * 0.5f, 0.0f);
    o.y = fmaxf((t.y + ar.y / cr + af.y / cf) * 0.5f, 0.0f);
    o.z = fmaxf((t.z + ar.z / cr + af.z / cf) * 0.5f, 0.0f);
    o.w = fmaxf((t.w + ar.w / cr + af.w / cf) * 0.5f, 0.0f);
    ((float4*)out_user)[g] = o;
}

__global__ void __launch_bounds__(256)
combine_item_kernel(const float* __restrict__ tgt_item,
                    const float* __restrict__ agg_rates,
                    const float* __restrict__ cnt_rates,
                    float* __restrict__ out_item, int n)
{
    const int g = blockIdx.x * blockDim.x + threadIdx.x;
    if (g >= n * 32) return;
    const int node = g >> 5;
    const float cr = fmaxf(cnt_rates[node], 1.0f);
    const float4 t  = ((const float4*)tgt_item)[g];
    const float4 ar = ((const float4*)agg_rates)[g];
    float4 o;
    o.x = fmaxf(t.x + ar.x / cr, 0.0f);
    o.y = fmaxf(t.y + ar.y / cr, 0.0f);
    o.z = fmaxf(t.z + ar.z / cr, 0.0f);
    o.w = fmaxf(t.w + ar.w / cr, 0.0f);
    ((float4*)out_item)[g] = o;
}

// ---------------------------------------------------------------------------
// Host-side launch
// ---------------------------------------------------------------------------
extern "C" void kernel_launch(void* const* d_in, const int* in_sizes, int n_in,
                              void* d_out, int out_size, void* d_ws, size_t ws_size,
                              hipStream_t stream)
{
    const float* x_user        = (const float*)d_in[0];
    const float* x_item        = (const float*)d_in[1];
    const float* w_rates_src   = (const float*)d_in[2];
    const float* w_rates_tgt   = (const float*)d_in[3];
    const float* w_rated_src   = (const float*)d_in[4];
    const float* w_rated_tgt   = (const float*)d_in[5];
    const float* w_follows_src = (const float*)d_in[6];
    const float* w_follows_tgt = (const float*)d_in[7];
    const int*   e_rates       = (const int*)d_in[8];    // [2, E]
    const int*   e_rated       = (const int*)d_in[9];
    const int*   e_follows     = (const int*)d_in[10];

    const int NU = N_NODES, NI = N_NODES, E = N_EDGES;
    const size_t NF = (size_t)N_NODES * DIMH;

    // Workspace layout (floats). Zeroed region (agg_* + cnt_*) is contiguous.
    float* ws          = (float*)d_ws;
    float* src_rates   = ws;                  // [NF]
    float* src_rated   = src_rates   + NF;    // [NF]
    float* src_follows = src_rated   + NF;    // [NF]
    float* tgt_item    = src_follows + NF;    // [NF]
    float* tgt_user    = tgt_item    + NF;    // [NF]
    float* agg_rates   = tgt_user    + NF;    // [NF]  (zeroed)
    float* agg_rated   = agg_rates   + NF;    // [NF]  (zeroed)
    float* agg_follows = agg_rated   + NF;    // [NF]  (zeroed)
    float* cnt_rates   = agg_follows + NF;    // [N]   (zeroed)
    float* cnt_rated   = cnt_rates   + N_NODES;
    float* cnt_follows = cnt_rated   + N_NODES;

    // Zero the atomic accumulation buffers + counts.
    hipMemsetAsync(agg_rates, 0,
                   (3 * NF + 3 * (size_t)N_NODES) * sizeof(float), stream);

    // Projections (6 GEMMs). tgt_user = x_user@w_rated_tgt + x_user@w_follows_tgt.
    const dim3 gblk(256);
    const dim3 ggrd_u((NU + 127) / 128);
    const dim3 ggrd_i((NI + 127) / 128);
    gemm_proj_kernel<<<ggrd_u, gblk, 0, stream>>>(x_user, w_rates_src,   src_rates,   NU, 0);
    gemm_proj_kernel<<<ggrd_i, gblk, 0, stream>>>(x_item, w_rated_src,   src_rated,   NI, 0);
    gemm_proj_kernel<<<ggrd_u, gblk, 0, stream>>>(x_user, w_follows_src, src_follows, NU, 0);
    gemm_proj_kernel<<<ggrd_i, gblk, 0, stream>>>(x_item, w_rates_tgt,   tgt_item,    NI, 0);
    gemm_proj_kernel<<<ggrd_u, gblk, 0, stream>>>(x_user, w_rated_tgt,   tgt_user,    NU, 0);
    gemm_proj_kernel<<<ggrd_u, gblk, 0, stream>>>(x_user, w_follows_tgt, tgt_user,    NU, 1);

    // Scatter-mean numerators + counts. Edge arrays are [2,E]: row0=src, row1=tgt.
    const int sc_blocks = (E * 32 + 255) / 256;
    scatter_add_kernel<<<sc_blocks, 256, 0, stream>>>(src_rates,   e_rates,   e_rates   + E, E, agg_rates,   cnt_rates);
    scatter_add_kernel<<<sc_blocks, 256, 0, stream>>>(src_rated,   e_rated,   e_rated   + E, E, agg_rated,   cnt_rated);
    scatter_add_kernel<<<sc_blocks, 256, 0, stream>>>(src_follows, e_follows, e_follows + E, E, agg_follows, cnt_follows);

    // Combine + ReLU. d_out = [out_user (NF) | out_item (NF)].
    float* out_user = (float*)d_out;
    float* out_item = out_user + NF;
    const int cb_blocks = (N_NODES * 32 + 255) / 256;
    combine_user_kernel<<<cb_blocks, 256, 0, stream>>>(tgt_user, agg_rated, cnt_rated,
                                                       agg_follows, cnt_follows,
                                                       out_user, NU);
    combine_item_kernel<<<cb_blocks, 256, 0, stream>>>(tgt_item, agg_rates, cnt_rates,
                                                       out_item, NI);
}